// TransformerDecoderCross_45397804319188
// MI455X (gfx1250) — compile-verified
//
#include <hip/hip_runtime.h>
#include <hip/hip_bf16.h>

// Problem constants (match reference)
#define N_TOK 1024
#define M_TOK 512
#define DIM   128
#define NHEAD 4
#define HDIM  32
#define NLAYER 2
#define FFDIM 256

typedef __attribute__((ext_vector_type(2))) float v2f;
typedef __attribute__((ext_vector_type(8))) float v8f;

// Native fp32 matrix op: D(16x16) += A(16x4) * B(4x16), RNE accumulate.
__device__ __forceinline__ v8f wmma4(v2f a, v2f b, v8f c) {
  return __builtin_amdgcn_wmma_f32_16x16x4_f32(
      /*neg_a=*/false, a, /*neg_b=*/false, b,
      /*c_mod=*/(short)0, c, /*reuse_a=*/false, /*reuse_b=*/false);
}

// Async global -> LDS copy (CDNA5 GLOBAL_LOAD_ASYNC_TO_LDS_B32, ASYNCcnt).
// Low 32 bits of a flat shared-aperture pointer are the LDS byte offset.
__device__ __forceinline__ void g2lds_b32(void* lds, const void* g) {
  unsigned loff = (unsigned)(uintptr_t)lds;
  asm volatile("global_load_async_to_lds_b32 %0, %1, off"
               :: "v"(loff), "v"(g) : "memory");
}
__device__ __forceinline__ void wait_async0() {
  asm volatile("s_wait_asynccnt 0x0" ::: "memory");
}

// ---------------------------------------------------------------------------
// Generic GEMM: C[rows x cols] = A[rows x K] * B[K x cols] + bias (+ReLU).
// One wave32 per 16x64 output strip: 4 accumulators share one A fragment per
// K-step -> 4 back-to-back WMMAs per 1 A-load, 4x less A traffic.
// grid = (cols/64, rows/16).
// ---------------------------------------------------------------------------
__global__ __launch_bounds__(32)
void k_gemm(const float* __restrict__ A, int lda,
            const float* __restrict__ B, int ldb,
            const float* __restrict__ bias,
            float* __restrict__ C, int ldc,
            int K, int relu)
{
  const int lane = threadIdx.x;
  const int half = lane >> 4, l16 = lane & 15;
  const int row0 = blockIdx.y * 16, col0 = blockIdx.x * 64;

  const float* Ar = A + (size_t)(row0 + l16) * lda + half * 2;
  const float* Bc = B + (size_t)(half * 2) * ldb + col0 + l16;

  v8f acc0 = {}, acc1 = {}, acc2 = {}, acc3 = {};
  for (int kb = 0; kb < K; kb += 4) {
    v2f a;
    a.x = Ar[kb];
    a.y = Ar[kb + 1];
    const float* bp = Bc + (size_t)kb * ldb;
    v2f b0, b1, b2, b3;
    b0.x = bp[0];   b0.y = bp[(size_t)ldb + 0];
    b1.x = bp[16];  b1.y = bp[(size_t)ldb + 16];
    b2.x = bp[32];  b2.y = bp[(size_t)ldb + 32];
    b3.x = bp[48];  b3.y = bp[(size_t)ldb + 48];
    acc0 = wmma4(a, b0, acc0);
    acc1 = wmma4(a, b1, acc1);
    acc2 = wmma4(a, b2, acc2);
    acc3 = wmma4(a, b3, acc3);
  }

  const float bv0 = bias ? bias[col0 + l16]      : 0.0f;
  const float bv1 = bias ? bias[col0 + 16 + l16] : 0.0f;
  const float bv2 = bias ? bias[col0 + 32 + l16] : 0.0f;
  const float bv3 = bias ? bias[col0 + 48 + l16] : 0.0f;
#pragma unroll
  for (int v = 0; v < 8; ++v) {
    float* crow = C + (size_t)(row0 + half * 8 + v) * ldc + col0 + l16;
    float v0 = acc0[v] + bv0, v1 = acc1[v] + bv1;
    float v2 = acc2[v] + bv2, v3 = acc3[v] + bv3;
    if (relu) {
      v0 = fmaxf(v0, 0.0f); v1 = fmaxf(v1, 0.0f);
      v2 = fmaxf(v2, 0.0f); v3 = fmaxf(v3, 0.0f);
    }
    crow[0] = v0; crow[16] = v1; crow[32] = v2; crow[48] = v3;
  }
}

// ---------------------------------------------------------------------------
// Attention scores: S[h,q,k] = scale * sum_d Q[q, h*32+d] * K[k, h*32+d]
// One wave per 16q x 64k strip (4 accumulators share the Q fragment).
// grid = (Nk/64, Nq/16, NHEAD). K operand read "transposed" from memory.
// ---------------------------------------------------------------------------
__global__ __launch_bounds__(32)
void k_attn_scores(const float* __restrict__ Q, int ldq,
                   const float* __restrict__ Kb, int ldk,
                   float* __restrict__ S, int Nk, float scale)
{
  const int lane = threadIdx.x;
  const int half = lane >> 4, l16 = lane & 15;
  const int col0 = blockIdx.x * 64;   // key tile base
  const int row0 = blockIdx.y * 16;   // query tile
  const int h    = blockIdx.z;
  const int Nq   = gridDim.y * 16;

  const float* Qr = Q  + (size_t)(row0 + l16) * ldq + h * HDIM + half * 2;
  const float* Kr = Kb + (size_t)(col0 + l16) * ldk + h * HDIM + half * 2;
  const size_t kstep = (size_t)16 * ldk;

  v8f acc0 = {}, acc1 = {}, acc2 = {}, acc3 = {};
#pragma unroll
  for (int kb = 0; kb < HDIM; kb += 4) {
    v2f a, b0, b1, b2, b3;
    a.x  = Qr[kb];              a.y  = Qr[kb + 1];
    b0.x = Kr[kb];              b0.y = Kr[kb + 1];
    b1.x = Kr[kstep + kb];      b1.y = Kr[kstep + kb + 1];
    b2.x = Kr[2 * kstep + kb];  b2.y = Kr[2 * kstep + kb + 1];
    b3.x = Kr[3 * kstep + kb];  b3.y = Kr[3 * kstep + kb + 1];
    acc0 = wmma4(a, b0, acc0);
    acc1 = wmma4(a, b1, acc1);
    acc2 = wmma4(a, b2, acc2);
    acc3 = wmma4(a, b3, acc3);
  }

  float* Sp = S + (size_t)h * Nq * Nk;
#pragma unroll
  for (int v = 0; v < 8; ++v) {
    float* srow = Sp + (size_t)(row0 + half * 8 + v) * Nk + col0 + l16;
    srow[0]  = acc0[v] * scale;
    srow[16] = acc1[v] * scale;
    srow[32] = acc2[v] * scale;
    srow[48] = acc3[v] * scale;
  }
}

// ---------------------------------------------------------------------------
// Attention output: O[q, h*32+d] = sum_k S[h,q,k] * V[k, h*32+d]
// One wave covers the full 32-wide head (2 accumulators share the S
// fragment) -> halves reads of the 16MB score buffer (largest stream).
// grid = (Nq/16, NHEAD). K dim = Nk (1024 self / 512 cross).
// ---------------------------------------------------------------------------
__global__ __launch_bounds__(32)
void k_attn_av(const float* __restrict__ S, int Nk,
               const float* __restrict__ V, int ldv,
               float* __restrict__ O, int ldo)
{
  const int lane = threadIdx.x;
  const int half = lane >> 4, l16 = lane & 15;
  const int row0 = blockIdx.x * 16;
  const int h    = blockIdx.y;
  const int Nq   = gridDim.x * 16;

  const float* Ar = S + (size_t)h * Nq * Nk + (size_t)(row0 + l16) * Nk + half * 2;
  const float* Bc = V + (size_t)(half * 2) * ldv + h * HDIM + l16;

  v8f acc0 = {}, acc1 = {};
  for (int kb = 0; kb < Nk; kb += 4) {
    v2f a, b0, b1;
    a.x = Ar[kb];  a.y = Ar[kb + 1];
    const float* bp = Bc + (size_t)kb * ldv;
    b0.x = bp[0];   b0.y = bp[(size_t)ldv + 0];
    b1.x = bp[16];  b1.y = bp[(size_t)ldv + 16];
    acc0 = wmma4(a, b0, acc0);
    acc1 = wmma4(a, b1, acc1);
  }

#pragma unroll
  for (int v = 0; v < 8; ++v) {
    float* orow = O + (size_t)(row0 + half * 8 + v) * ldo + h * HDIM + l16;
    orow[0]  = acc0[v];
    orow[16] = acc1[v];
  }
}

// ---------------------------------------------------------------------------
// Row softmax in place. grid = (Nq, NHEAD), block = 256.
// ---------------------------------------------------------------------------
__global__ __launch_bounds__(256)
void k_softmax(float* __restrict__ S, int Nk)
{
  __shared__ float red[256];
  const int row = blockIdx.x, h = blockIdx.y;
  const int Nq  = gridDim.x;
  float* p = S + ((size_t)h * Nq + row) * Nk;
  const int tid = threadIdx.x;

  float m = -1e30f;
  for (int i = tid; i < Nk; i += 256) m = fmaxf(m, p[i]);
  red[tid] = m; __syncthreads();
  for (int s = 128; s > 0; s >>= 1) {
    if (tid < s) red[tid] = fmaxf(red[tid], red[tid + s]);
    __syncthreads();
  }
  m = red[0]; __syncthreads();

  float sum = 0.0f;
  for (int i = tid; i < Nk; i += 256) {
    float e = __expf(p[i] - m);
    p[i] = e;
    sum += e;
  }
  red[tid] = sum; __syncthreads();
  for (int s = 128; s > 0; s >>= 1) {
    if (tid < s) red[tid] += red[tid + s];
    __syncthreads();
  }
  const float inv = 1.0f / red[0];
  for (int i = tid; i < Nk; i += 256) p[i] *= inv;
}

// ---------------------------------------------------------------------------
// y = LayerNorm(x + t) * g + b   (row-wise, D = 128 = blockDim)
// ---------------------------------------------------------------------------
__global__ __launch_bounds__(128)
void k_add_ln(const float* __restrict__ x, const float* __restrict__ t,
              const float* __restrict__ g, const float* __restrict__ b,
              float* __restrict__ y)
{
  __shared__ float red[128];
  const int row = blockIdx.x, tid = threadIdx.x;
  const float v = x[(size_t)row * DIM + tid] + t[(size_t)row * DIM + tid];

  red[tid] = v; __syncthreads();
  for (int s = 64; s > 0; s >>= 1) {
    if (tid < s) red[tid] += red[tid + s];
    __syncthreads();
  }
  const float mu = red[0] * (1.0f / DIM); __syncthreads();

  const float d = v - mu;
  red[tid] = d * d; __syncthreads();
  for (int s = 64; s > 0; s >>= 1) {
    if (tid < s) red[tid] += red[tid + s];
    __syncthreads();
  }
  const float var = red[0] * (1.0f / DIM);
  y[(size_t)row * DIM + tid] = d * rsqrtf(var + 1e-5f) * g[tid] + b[tid];
}

// ---------------------------------------------------------------------------
// x = concat(feat[N,11], pe[N,8]) @ Wp[19,D] + bp.  K=19: plain VALU.
// ---------------------------------------------------------------------------
__global__ void k_input_proj(const float* __restrict__ feat, const float* __restrict__ pe,
                             const float* __restrict__ Wp, const float* __restrict__ bp,
                             float* __restrict__ x)
{
  const int idx = blockIdx.x * blockDim.x + threadIdx.x;
  if (idx >= N_TOK * DIM) return;
  const int n = idx / DIM, d = idx % DIM;
  float s = bp[d];
#pragma unroll
  for (int k = 0; k < 11; ++k) s += feat[n * 11 + k] * Wp[k * DIM + d];
#pragma unroll
  for (int k = 0; k < 8; ++k)  s += pe[n * 8 + k]   * Wp[(11 + k) * DIM + d];
  x[idx] = s;
}

// ---------------------------------------------------------------------------
// Edge head: out[i,j] = (i==j) ? 0 : sigmoid(sum_d relu(ai[i,d]+bj[j,d])*w2[d] + be2)
// ReLU inside the contraction -> not a GEMM; LDS-tiled VALU (32x32 pair tile).
// Tiles staged with async global->LDS copies (ASYNCcnt path, no VGPR bounce).
// be1 is pre-folded into ai by its GEMM bias. grid=(N/32, N/32), block=256.
// ---------------------------------------------------------------------------
__global__ __launch_bounds__(256)
void k_edge(const float* __restrict__ ai, const float* __restrict__ bj,
            const float* __restrict__ w2, const float* __restrict__ be2,
            float* __restrict__ out)
{
  __shared__ float sa[32][DIM];       // read as wave-broadcast: no pad needed
  __shared__ float sb[32][DIM + 1];   // padded: stride 129 avoids bank conflicts
  __shared__ float sw[DIM];
  const int i0 = blockIdx.y * 32, j0 = blockIdx.x * 32;
  const int tid = threadIdx.x;

  for (int t = tid; t < 32 * DIM; t += 256) {
    const int r = t / DIM, c = t % DIM;
    g2lds_b32(&sa[r][c], &ai[(size_t)(i0 + r) * DIM + c]);
    g2lds_b32(&sb[r][c], &bj[(size_t)(j0 + r) * DIM + c]);
  }
  if (tid < DIM) g2lds_b32(&sw[tid], &w2[tid]);
  wait_async0();
  __syncthreads();

  const float b2 = be2[0];
  const int jj = tid & 31, ty = tid >> 5;   // 8 waves x 4 rows each
#pragma unroll
  for (int q = 0; q < 4; ++q) {
    const int ii = ty + q * 8;
    float acc = 0.0f;
    for (int d = 0; d < DIM; ++d) {
      const float z = sa[ii][d] + sb[jj][d];
      acc += fmaxf(z, 0.0f) * sw[d];
    }
    const float logit = acc + b2;
    const float pv = 1.0f / (1.0f + __expf(-logit));
    const int gi = i0 + ii, gj = j0 + jj;
    out[(size_t)gi * N_TOK + gj] = (gi == gj) ? 0.0f : pv;
  }
}

// ---------------------------------------------------------------------------
// Host orchestration.
// ---------------------------------------------------------------------------
extern "C" void kernel_launch(void* const* d_in, const int* in_sizes, int n_in,
                              void* d_out, int out_size, void* d_ws, size_t ws_size,
                              hipStream_t stream)
{
  const float* mask_nodes = (const float*)d_in[0];
  const float* sincos_pe  = (const float*)d_in[1];
  const float* h_pool     = (const float*)d_in[2];
  const float* Wp = (const float*)d_in[3];
  const float* bp = (const float*)d_in[4];
  const float* sa_qkv_w = (const float*)d_in[5];
  const float* sa_qkv_b = (const float*)d_in[6];
  const float* sa_out_w = (const float*)d_in[7];
  const float* sa_out_b = (const float*)d_in[8];
  const float* ca_qkv_w = (const float*)d_in[9];
  const float* ca_qkv_b = (const float*)d_in[10];
  const float* ca_out_w = (const float*)d_in[11];
  const float* ca_out_b = (const float*)d_in[12];
  const float* ff1_w = (const float*)d_in[13];
  const float* ff1_b = (const float*)d_in[14];
  const float* ff2_w = (const float*)d_in[15];
  const float* ff2_b = (const float*)d_in[16];
  const float* ln1_g = (const float*)d_in[17];
  const float* ln1_b = (const float*)d_in[18];
  const float* ln2_g = (const float*)d_in[19];
  const float* ln2_b = (const float*)d_in[20];
  const float* ln3_g = (const float*)d_in[21];
  const float* ln3_b = (const float*)d_in[22];
  const float* We1 = (const float*)d_in[23];
  const float* be1 = (const float*)d_in[24];
  const float* We2 = (const float*)d_in[25];
  const float* be2 = (const float*)d_in[26];
  float* out = (float*)d_out;

  // Workspace carve-up (~22 MB total; scores scratch dominates, fits L2)
  float* ws = (float*)d_ws;
  float* x0   = ws; ws += N_TOK * DIM;
  float* x1   = ws; ws += N_TOK * DIM;
  float* qkv  = ws; ws += N_TOK * 3 * DIM;
  float* kv   = ws; ws += M_TOK * 2 * DIM;
  float* qb   = ws; ws += N_TOK * DIM;
  float* ob   = ws; ws += N_TOK * DIM;
  float* tb   = ws; ws += N_TOK * DIM;
  float* ffb  = ws; ws += N_TOK * FFDIM;
  float* aibuf = ws; ws += N_TOK * DIM;
  float* bjbuf = ws; ws += N_TOK * DIM;
  float* S    = ws; ws += (size_t)NHEAD * N_TOK * N_TOK;

  const float scale = 0.17677669529663687f;  // 1/sqrt(32)

  k_input_proj<<<(N_TOK * DIM + 255) / 256, 256, 0, stream>>>(
      mask_nodes, sincos_pe, Wp, bp, x0);

  float* x  = x0;
  float* xn = x1;
  for (int l = 0; l < NLAYER; ++l) {
    // ---- self-attention ----
    k_gemm<<<dim3(3 * DIM / 64, N_TOK / 16), 32, 0, stream>>>(
        x, DIM, sa_qkv_w + (size_t)l * DIM * 3 * DIM, 3 * DIM,
        sa_qkv_b + l * 3 * DIM, qkv, 3 * DIM, DIM, 0);
    k_attn_scores<<<dim3(N_TOK / 64, N_TOK / 16, NHEAD), 32, 0, stream>>>(
        qkv, 3 * DIM, qkv + DIM, 3 * DIM, S, N_TOK, scale);
    k_softmax<<<dim3(N_TOK, NHEAD), 256, 0, stream>>>(S, N_TOK);
    k_attn_av<<<dim3(N_TOK / 16, NHEAD), 32, 0, stream>>>(
        S, N_TOK, qkv + 2 * DIM, 3 * DIM, ob, DIM);
    k_gemm<<<dim3(DIM / 64, N_TOK / 16), 32, 0, stream>>>(
        ob, DIM, sa_out_w + (size_t)l * DIM * DIM, DIM,
        sa_out_b + l * DIM, tb, DIM, DIM, 0);
    k_add_ln<<<N_TOK, 128, 0, stream>>>(x, tb, ln1_g + l * DIM, ln1_b + l * DIM, xn);
    { float* t_ = x; x = xn; xn = t_; }

    // ---- cross-attention (kv from h_pool) ----
    k_gemm<<<dim3(DIM / 64, N_TOK / 16), 32, 0, stream>>>(
        x, DIM, ca_qkv_w + (size_t)l * DIM * 3 * DIM, 3 * DIM,
        ca_qkv_b + l * 3 * DIM, qb, DIM, DIM, 0);
    k_gemm<<<dim3(2 * DIM / 64, M_TOK / 16), 32, 0, stream>>>(
        h_pool, DIM, ca_qkv_w + (size_t)l * DIM * 3 * DIM + DIM, 3 * DIM,
        ca_qkv_b + l * 3 * DIM + DIM, kv, 2 * DIM, DIM, 0);
    k_attn_scores<<<dim3(M_TOK / 64, N_TOK / 16, NHEAD), 32, 0, stream>>>(
        qb, DIM, kv, 2 * DIM, S, M_TOK, scale);
    k_softmax<<<dim3(N_TOK, NHEAD), 256, 0, stream>>>(S, M_TOK);
    k_attn_av<<<dim3(N_TOK / 16, NHEAD), 32, 0, stream>>>(
        S, M_TOK, kv + DIM, 2 * DIM, ob, DIM);
    k_gemm<<<dim3(DIM / 64, N_TOK / 16), 32, 0, stream>>>(
        ob, DIM, ca_out_w + (size_t)l * DIM * DIM, DIM,
        ca_out_b + l * DIM, tb, DIM, DIM, 0);
    k_add_ln<<<N_TOK, 128, 0, stream>>>(x, tb, ln2_g + l * DIM, ln2_b + l * DIM, xn);
    { float* t_ = x; x = xn; xn = t_; }

    // ---- feed-forward ----
    k_gemm<<<dim3(FFDIM / 64, N_TOK / 16), 32, 0, stream>>>(
        x, DIM, ff1_w + (size_t)l * DIM * FFDIM, FFDIM,
        ff1_b + l * FFDIM, ffb, FFDIM, DIM, 1);
    k_gemm<<<dim3(DIM / 64, N_TOK / 16), 32, 0, stream>>>(
        ffb, FFDIM, ff2_w + (size_t)l * FFDIM * DIM, DIM,
        ff2_b + l * DIM, tb, DIM, FFDIM, 0);
    k_add_ln<<<N_TOK, 128, 0, stream>>>(x, tb, ln3_g + l * DIM, ln3_b + l * DIM, xn);
    { float* t_ = x; x = xn; xn = t_; }
  }

  // ---- pairwise edge head ----
  k_gemm<<<dim3(DIM / 64, N_TOK / 16), 32, 0, stream>>>(
      x, DIM, We1, DIM, be1, aibuf, DIM, DIM, 0);              // be1 folded here
  k_gemm<<<dim3(DIM / 64, N_TOK / 16), 32, 0, stream>>>(
      x, DIM, We1 + DIM * DIM, DIM, nullptr, bjbuf, DIM, DIM, 0);
  k_edge<<<dim3(N_TOK / 32, N_TOK / 32), 256, 0, stream>>>(
      aibuf, bjbuf, We2, be2, out);

  (void)in_sizes; (void)n_in; (void)out_size; (void)ws_size;
}